// RotMap_54047868452925
// MI455X (gfx1250) — compile-verified
//
#include <hip/hip_runtime.h>

// RotMap: Y[n,a,b,i,f] = sum_c w[a,c,i] * x[n,c,b,i,f]
// Memory-bound streaming kernel (0.75 FLOP/byte); 3x3 rotation routed through
// V_WMMA_F32_16X16X4_F32 (A = padded W_i 16x4, B = 16 x-columns 4x16).

typedef float v2f __attribute__((ext_vector_type(2)));
typedef float v8f __attribute__((ext_vector_type(8)));

#define N_BATCH 256
#define D_INNER 3
#define NUM_ROT 342
#define FRAME 100

// x / y layout: [n, c, b, i, f] row-major, element strides:
#define S_I  FRAME                        // 100
#define S_B  (NUM_ROT * FRAME)            // 34200
#define S_C  (D_INNER * S_B)              // 102600
#define S_N  (D_INNER * S_C)              // 307800

#define COLS_PER_I      (N_BATCH * D_INNER * FRAME)   // 76800 (n,b,f) columns per i
#define GROUPS_PER_WAVE 8
#define COLS_PER_WAVE   (16 * GROUPS_PER_WAVE)        // 128
#define WAVES_PER_I     (COLS_PER_I / COLS_PER_WAVE)  // 600 (exact)
#define WAVES_PER_BLOCK 8
#define TOTAL_WAVES     (NUM_ROT * WAVES_PER_I)       // 205200
#define GRID_BLOCKS     (TOTAL_WAVES / WAVES_PER_BLOCK) // 25650 (exact)

__global__ __launch_bounds__(256)
void RotMap_wmma_f32_kernel(const float* __restrict__ x,
                            const float* __restrict__ w,
                            float* __restrict__ y) {
    const unsigned lane = threadIdx.x & 31u;
    const unsigned half = lane >> 4;        // 0: lanes 0-15, 1: lanes 16-31
    const unsigned l    = lane & 15u;

    const unsigned waveInBlock = threadIdx.x >> 5;
    const unsigned waveId = blockIdx.x * WAVES_PER_BLOCK + waveInBlock;
    const unsigned i      = waveId / WAVES_PER_I;   // rotation index
    const unsigned chunk  = waveId % WAVES_PER_I;

    // ---- A-matrix: W_i padded into 16x4 (rows 0-2, cols 0-2 live) ----
    // Layout (32-bit A 16x4): lanes 0-15 -> M=lane, VGPR0=K0, VGPR1=K1;
    //                         lanes 16-31 -> same M,  VGPR0=K2, VGPR1=K3.
    v2f A;
    {
        float a0 = 0.0f, a1 = 0.0f;
        if (l < D_INNER) {
            const unsigned c_lo = half ? 2u : 0u;               // K0 / K2
            a0 = w[(l * D_INNER + c_lo) * NUM_ROT + i];
            if (!half)                                          // K1 (K3 = 0 pad)
                a1 = w[(l * D_INNER + 1u) * NUM_ROT + i];
        }
        A.x = a0;
        A.y = a1;
    }

    const unsigned colBase = chunk * COLS_PER_WAVE + l;

#pragma unroll
    for (int g = 0; g < GROUPS_PER_WAVE; ++g) {
        // Flat (n,b,f) column this lane owns for this WMMA group.
        const unsigned col = colBase + (unsigned)g * 16u;
        const unsigned nb  = col / FRAME;
        const unsigned f   = col % FRAME;
        const unsigned n   = nb / D_INNER;
        const unsigned b   = nb % D_INNER;
        const unsigned off0 = n * S_N + b * S_B + i * S_I + f;  // c = 0 element

        // ---- B-matrix: 4x16 column block ----
        // lanes 0-15: VGPR0 = c0 row, VGPR1 = c1 row
        // lanes 16-31: VGPR0 = c2 row, VGPR1 = zero pad (K=3)
        v2f B;
        B.x = x[off0 + (half ? 2u * S_C : 0u)];
        float by = 0.0f;
        if (!half)
            by = x[off0 + S_C];
        B.y = by;

        // WMMA with full EXEC (divergence above is closed before this point).
        v8f C = {};
        C = __builtin_amdgcn_wmma_f32_16x16x4_f32(
                /*neg_a=*/false, A, /*neg_b=*/false, B,
                /*c_mod=*/(short)0, C, /*reuse_a=*/false, /*reuse_b=*/false);

        // ---- D: rows 0..2 (lanes 0-15, VGPRs 0..2) are the rotated column ----
        if (!half) {
            y[off0]            = C[0];
            y[off0 + S_C]      = C[1];
            y[off0 + 2u * S_C] = C[2];
        }
    }
}

extern "C" void kernel_launch(void* const* d_in, const int* in_sizes, int n_in,
                              void* d_out, int out_size, void* d_ws, size_t ws_size,
                              hipStream_t stream) {
    (void)in_sizes; (void)n_in; (void)d_ws; (void)ws_size; (void)out_size;
    const float* x = (const float*)d_in[0];
    const float* w = (const float*)d_in[1];
    float*       y = (float*)d_out;

    dim3 grid(GRID_BLOCKS);
    dim3 block(WAVES_PER_BLOCK * 32);
    RotMap_wmma_f32_kernel<<<grid, block, 0, stream>>>(x, w, y);
}